// GNNEncoder_4398046511761
// MI455X (gfx1250) — compile-verified
//
#include <hip/hip_runtime.h>
#include <cstdint>

typedef __attribute__((ext_vector_type(2))) float v2f;
typedef __attribute__((ext_vector_type(8))) float v8f;

#define GCN_F_IN   512
#define GCN_F_HID  256
#define GCN_BN_EPS 1e-5f
#define GCN_SLOPE  0.01f

#define GEMM_BK    32          // K-chunk staged in LDS per iteration
#define GEMM_APITCH 44         // A-tile LDS row pitch (floats): 44*m mod 64 is a
                               // permutation for m=0..15 -> conflict-free fragment
                               // reads; 44*4B=176B keeps b128 stores 16B-aligned

// ---------------- degree / normalization ----------------

__global__ __launch_bounds__(256) void k_deg_init(float* __restrict__ deg, int n) {
  int i = blockIdx.x * 256 + threadIdx.x;
  if (i < n) deg[i] = 1.0f;  // self-loop contributes 1 to every node's degree
}

__global__ __launch_bounds__(256) void k_deg_count(const int* __restrict__ dst,
                                                   float* __restrict__ deg, int e) {
  int i = blockIdx.x * 256 + threadIdx.x;
  if (i < e) atomicAdd(&deg[dst[i]], 1.0f);
}

__global__ __launch_bounds__(256) void k_deg_rsqrt(float* __restrict__ deg, int n) {
  int i = blockIdx.x * 256 + threadIdx.x;
  if (i < n) deg[i] = rsqrtf(deg[i]);  // deg >= 1 always (self-loop)
}

// ---------------- fp32 WMMA GEMM:  T[n,256] = X[n,KDIM] @ W[KDIM,256] ----------------
// Block = 128 threads = 4 waves; block covers 16 rows x 256 cols.
// A-tiles (16 x 32 f32) are double-buffered in LDS, filled with
// GLOBAL_LOAD_ASYNC_TO_LDS_B128 (ASYNCcnt) so the fill is coalesced, bypasses
// VGPRs, and overlaps the WMMAs of the current tile. Fragments come from LDS
// (ds_load_b64); B fragments come straight from global (W is <=512 KB, L2/WGP$
// resident). Matrix op: V_WMMA_F32_16X16X4_F32 -- exact fp32, matches reference.
//
// A (16x4 f32) lane layout: lanes 0-15 hold K=0(v0),1(v1); lanes 16-31 hold K=2,3.
// B (4x16 f32) mirrors: lanes 0-15 -> K=0,1 ; lanes 16-31 -> K=2,3 ; N = lane&15.
// C/D (16x16 f32): VGPR j -> M=j (lanes 0-15) / M=8+j (lanes 16-31), N = lane&15.

template <int KDIM>
__global__ __launch_bounds__(128) void k_gemm_wmma(const float* __restrict__ X,
                                                   const float* __restrict__ W,
                                                   float* __restrict__ T, int n) {
  __shared__ float As[2][16 * GEMM_APITCH];

  const int tid  = threadIdx.x;
  const int lane = tid & 31;
  const int wave = tid >> 5;
  const int m    = lane & 15;
  const int koff = (lane >> 4) << 1;       // 0 or 2
  const int row0 = blockIdx.x << 4;
  const int colBase = wave << 6;           // 0,64,128,192

  // cooperative A staging: thread t fills As[buf][(t>>3)*APITCH + (t&7)*4 ..+3]
  // from X[row0 + (t>>3)][kc*BK + (t&7)*4 ..+3]  (one async b128 per thread)
  const int ar = tid >> 3;                 // 0..15 (tile row)
  const int ac = (tid & 7) << 2;           // 0,4,...,28 (tile col)
  int arow = row0 + ar;
  if (arow >= n) arow = n - 1;             // clamp loads; stores are guarded
  const float* aSrc = X + (size_t)arow * KDIM + ac;
  // LDS byte address = low 32 bits of the generic pointer (ISA 10.2: LDS
  // aperture addresses truncate to addr[31:0])
  const unsigned ldsDst0 = (unsigned)(uintptr_t)&As[0][ar * GEMM_APITCH + ac];
  const unsigned ldsDst1 = (unsigned)(uintptr_t)&As[1][ar * GEMM_APITCH + ac];

  // preload tile 0
  asm volatile("global_load_async_to_lds_b128 %0, %1, off"
               :: "v"(ldsDst0), "v"(aSrc) : "memory");
  asm volatile("s_wait_asynccnt 0" ::: "memory");
  __syncthreads();

  v8f acc0 = {}, acc1 = {}, acc2 = {}, acc3 = {};

  const int NK = KDIM / GEMM_BK;
  for (int kc = 0; kc < NK; ++kc) {
    // kick off the fill of the other buffer while we compute on this one
    if (kc + 1 < NK) {
      const float* nxt = aSrc + (size_t)(kc + 1) * GEMM_BK;
      unsigned dst = ((kc + 1) & 1) ? ldsDst1 : ldsDst0;
      asm volatile("global_load_async_to_lds_b128 %0, %1, off"
                   :: "v"(dst), "v"(nxt) : "memory");
    }

    const float* aTile = &As[kc & 1][0];
    const int kbase = kc * GEMM_BK;
#pragma unroll
    for (int kk = 0; kk < GEMM_BK; kk += 4) {
      v2f a = *(const v2f*)&aTile[m * GEMM_APITCH + kk + koff];  // ds_load_b64

      const float* wr0 = W + (size_t)(kbase + kk + koff) * GCN_F_HID + colBase + m;
      const float* wr1 = wr0 + GCN_F_HID;
      v2f b0; b0.x = wr0[0];  b0.y = wr1[0];
      v2f b1; b1.x = wr0[16]; b1.y = wr1[16];
      v2f b2; b2.x = wr0[32]; b2.y = wr1[32];
      v2f b3; b3.x = wr0[48]; b3.y = wr1[48];

      acc0 = __builtin_amdgcn_wmma_f32_16x16x4_f32(false, a, false, b0, (short)0, acc0, false, false);
      acc1 = __builtin_amdgcn_wmma_f32_16x16x4_f32(false, a, false, b1, (short)0, acc1, false, false);
      acc2 = __builtin_amdgcn_wmma_f32_16x16x4_f32(false, a, false, b2, (short)0, acc2, false, false);
      acc3 = __builtin_amdgcn_wmma_f32_16x16x4_f32(false, a, false, b3, (short)0, acc3, false, false);
    }

    // next buffer must be fully in LDS, and everyone done reading, before reuse
    asm volatile("s_wait_asynccnt 0" ::: "memory");
    __syncthreads();
  }

  const int nOff  = lane & 15;
  const int mBase = row0 + ((lane >> 4) << 3);  // lanes 16-31 start at M=8
#pragma unroll
  for (int j = 0; j < 8; ++j) {
    int rr = mBase + j;
    if (rr < n) {
      float* trow = T + (size_t)rr * GCN_F_HID + colBase + nOff;
      trow[0]  = acc0[j];
      trow[16] = acc1[j];
      trow[32] = acc2[j];
      trow[48] = acc3[j];
    }
  }
}

// ---------------- self-loop + bias init of aggregation buffer ----------------
// agg[i,f] = T[i,f] * dinv[i]^2 + bias[f]   (segment_sum self-loop term + bias)

__global__ __launch_bounds__(256) void k_selfloop_bias(const float* __restrict__ T,
                                                       float* __restrict__ agg,
                                                       const float* __restrict__ dinv,
                                                       const float* __restrict__ bias,
                                                       long long total) {
  long long i = (long long)blockIdx.x * 256 + threadIdx.x;
  if (i >= total) return;
  int node = (int)(i >> 8);
  int f    = (int)(i & 255);
  float di = dinv[node];
  agg[i] = T[i] * di * di + bias[f];
}

// ---------------- edge scatter: agg[dst] += T[src] * dinv[src]*dinv[dst] ----------------
// 64 lanes per edge, float4 gather (L2-resident: T is 51 MB < 192 MB L2),
// atomicAdd -> GLOBAL_ATOMIC_ADD_F32 resolved at L2.

__global__ __launch_bounds__(256) void k_scatter(const float* __restrict__ T,
                                                 float* __restrict__ agg,
                                                 const int* __restrict__ src,
                                                 const int* __restrict__ dst,
                                                 const float* __restrict__ dinv, int e) {
  int eid = blockIdx.x * 4 + (threadIdx.x >> 6);
  if (eid >= e) return;
  int t = threadIdx.x & 63;
  int s = src[eid], d = dst[eid];
  float nrm = dinv[s] * dinv[d];
  const float4* ts = (const float4*)(T + (size_t)s * GCN_F_HID);
  float4 v = ts[t];
  float* ad = agg + (size_t)d * GCN_F_HID + (t << 2);
  atomicAdd(ad + 0, v.x * nrm);
  atomicAdd(ad + 1, v.y * nrm);
  atomicAdd(ad + 2, v.z * nrm);
  atomicAdd(ad + 3, v.w * nrm);
}

// ---------------- fused BN(eval) + LeakyReLU ----------------

__global__ __launch_bounds__(256) void k_bn_act(const float* __restrict__ a,
                                                float* __restrict__ h,
                                                const float* __restrict__ g,
                                                const float* __restrict__ bt,
                                                const float* __restrict__ rm,
                                                const float* __restrict__ rv,
                                                long long total) {
  long long i = (long long)blockIdx.x * 256 + threadIdx.x;
  if (i >= total) return;
  int f = (int)(i & 255);
  float sc = g[f] * rsqrtf(rv[f] + GCN_BN_EPS);
  float v  = (a[i] - rm[f]) * sc + bt[f];
  h[i] = v > 0.f ? v : v * GCN_SLOPE;
}

// layer 1: v = agg2 + h (residual, bias already in agg2); h2 = leaky(bn(v));
// then jump-knowledge: hgx <- h + h2  (overwrites h buffer in place, same index)

__global__ __launch_bounds__(256) void k_bn_act_res(const float* __restrict__ agg2,
                                                    float* __restrict__ hgx,
                                                    const float* __restrict__ g,
                                                    const float* __restrict__ bt,
                                                    const float* __restrict__ rm,
                                                    const float* __restrict__ rv,
                                                    long long total) {
  long long i = (long long)blockIdx.x * 256 + threadIdx.x;
  if (i >= total) return;
  int f = (int)(i & 255);
  float hv = hgx[i];
  float v  = agg2[i] + hv;
  float sc = g[f] * rsqrtf(rv[f] + GCN_BN_EPS);
  float t  = (v - rm[f]) * sc + bt[f];
  float h2 = t > 0.f ? t : t * GCN_SLOPE;
  hgx[i] = hv + h2;
}

// ---------------- launcher ----------------

extern "C" void kernel_launch(void* const* d_in, const int* in_sizes, int n_in,
                              void* d_out, int out_size, void* d_ws, size_t ws_size,
                              hipStream_t stream) {
  const float* x   = (const float*)d_in[0];
  const int*   ei  = (const int*)d_in[1];
  const float* W0  = (const float*)d_in[2];
  const float* b0  = (const float*)d_in[3];
  const float* W1  = (const float*)d_in[4];
  const float* b1  = (const float*)d_in[5];
  const float* W2  = (const float*)d_in[6];
  const float* b2  = (const float*)d_in[7];
  const float* g0  = (const float*)d_in[8];
  const float* bt0 = (const float*)d_in[9];
  const float* rm0 = (const float*)d_in[10];
  const float* rv0 = (const float*)d_in[11];
  const float* g1  = (const float*)d_in[12];
  const float* bt1 = (const float*)d_in[13];
  const float* rm1 = (const float*)d_in[14];
  const float* rv1 = (const float*)d_in[15];

  const int n = in_sizes[0] / GCN_F_IN;   // 50000
  const int e = in_sizes[1] / 2;          // 800000
  const int* src = ei;                    // edge_index[0]
  const int* dst = ei + e;                // edge_index[1]

  // workspace layout (floats): dinv[npad] | bufA[n*256] | bufB[n*256] | bufC[n*256]
  float* ws   = (float*)d_ws;
  size_t npad = ((size_t)n + 63) & ~(size_t)63;
  float* dinv = ws;
  float* bufA = ws + npad;                       // GEMM output T
  float* bufB = bufA + (size_t)n * GCN_F_HID;    // aggregation target
  float* bufC = bufB + (size_t)n * GCN_F_HID;    // h / global_x
  float* out  = (float*)d_out;

  const long long total = (long long)n * GCN_F_HID;
  const int gn   = (n + 255) / 256;
  const int ge   = (e + 255) / 256;
  const int gnf  = (int)((total + 255) / 256);
  const int gsc  = (e + 3) / 4;
  const int ggm  = (n + 15) / 16;

  // symmetric GCN normalization
  k_deg_init <<<gn, 256, 0, stream>>>(dinv, n);
  k_deg_count<<<ge, 256, 0, stream>>>(dst, dinv, e);
  k_deg_rsqrt<<<gn, 256, 0, stream>>>(dinv, n);

  // ---- layer 0 ----
  k_gemm_wmma<GCN_F_IN><<<ggm, 128, 0, stream>>>(x, W0, bufA, n);
  k_selfloop_bias<<<gnf, 256, 0, stream>>>(bufA, bufB, dinv, b0, total);
  k_scatter      <<<gsc, 256, 0, stream>>>(bufA, bufB, src, dst, dinv, e);
  k_bn_act       <<<gnf, 256, 0, stream>>>(bufB, bufC, g0, bt0, rm0, rv0, total); // bufC = h

  // ---- layer 1 (residual + jump knowledge) ----
  k_gemm_wmma<GCN_F_HID><<<ggm, 128, 0, stream>>>(bufC, W1, bufA, n);
  k_selfloop_bias<<<gnf, 256, 0, stream>>>(bufA, bufB, dinv, b1, total);
  k_scatter      <<<gsc, 256, 0, stream>>>(bufA, bufB, src, dst, dinv, e);
  k_bn_act_res   <<<gnf, 256, 0, stream>>>(bufB, bufC, g1, bt1, rm1, rv1, total); // bufC = h + h2

  // ---- layer 2 (aggregate straight into d_out) ----
  k_gemm_wmma<GCN_F_HID><<<ggm, 128, 0, stream>>>(bufC, W2, bufA, n);
  k_selfloop_bias<<<gnf, 256, 0, stream>>>(bufA, out, dinv, b2, total);
  k_scatter      <<<gsc, 256, 0, stream>>>(bufA, out, src, dst, dinv, e);

  (void)n_in; (void)out_size; (void)ws_size;
}